// SpatioTemporalAttention_51814485459215
// MI455X (gfx1250) — compile-verified
//
#include <hip/hip_runtime.h>
#include <hip/hip_bf16.h>

// SpatioTemporalAttention for MI455X (gfx1250, wave32).
// Dense GEMMs run on V_WMMA_F32_16X16X4_F32 (fp32-exact matrix path).
// Each wave computes a 64x32 block of C (4x2 grid of 16x16 WMMA tiles) so A/B
// fragments are register-reused: ~1 vector load per v_wmma instead of 3.
// Workspace requirement: ~110 MB.

namespace {
constexpr int N_    = 16;
constexpr int C_IN  = 128;
constexpr int T_    = 128;
constexpr int V_    = 25;
constexpr int EMB   = 256;
constexpr int HEADS = 8;
constexpr int WIN   = 5;
constexpr int DIL   = 2;
constexpr int PAD   = 4;          // (WIN-1)*DIL/2
constexpr int TV    = T_ * V_;    // 3200 (divisible by 32)
constexpr float INV_SQRT_HC = 0.17677669529663689f; // 1/sqrt(32)
constexpr float INV_WIN     = 0.2f;

constexpr int MT_W = 4;           // 16-row tiles per wave
constexpr int CT_W = 2;           // 16-col tiles per wave
} // namespace

typedef __attribute__((ext_vector_type(2))) float v2f;
typedef __attribute__((ext_vector_type(8))) float v8f;

static __device__ __forceinline__ int shift_t(int w, int t) {
  // reference _shift_idx semantics
  int widx = w - WIN / 2;
  int off  = (widx < 0 ? -widx : widx) * DIL;
  if (widx < 0)  { int s = t - off; return s < 0 ? 0 : s; }
  if (widx == 0) return t;
  return (t < T_ - off) ? t : (T_ - 1);
}

// ---------------------------------------------------------------------------
// GEMM: out[n, m, tv] = epilogue( sum_k W[m,k] * B[n,k,tv] )
// One wave computes a 64x32 block of C: 4x2 tiles of V_WMMA_F32_16X16X4_F32.
// MODE 0: out = acc + bias[m]                       (q, v convs)
// MODE 1: out = (acc + bias[m]) * pmul[n,m,t]       (k conv fused with *pq)
// MODE 2: out = acc/WIN + out  (out pre-holds sq/WIN + bkv partial)
// ---------------------------------------------------------------------------
template <int KDIM, int MODE>
__global__ __launch_bounds__(256)
void gemm_wmma_kernel(const float* __restrict__ W,
                      const float* __restrict__ bias,
                      const float* __restrict__ B,
                      const float* __restrict__ pmul,
                      float* __restrict__ out) {
  constexpr int MG = EMB / (16 * MT_W);  // 4  row groups
  constexpr int CG = TV / (16 * CT_W);   // 100 col groups
  const int wave = threadIdx.x >> 5;
  const int lane = threadIdx.x & 31;
  const int blk  = blockIdx.x * 8 + wave;   // grid sized exactly, no guard
  const int n    = blk / (MG * CG);
  const int rem  = blk % (MG * CG);
  const int m0   = (rem / CG) * (16 * MT_W);
  const int col0 = (rem % CG) * (16 * CT_W);

  const int half = lane >> 4;   // 0 -> K = k0,k0+1 ; 1 -> K = k0+2,k0+3
  const int l    = lane & 15;

  // A fragment source: row (m0 + mt*16 + l) of W, K offset 2*half
  const float* Wrow = W + (size_t)(m0 + l) * KDIM + 2 * half;
  // B fragment source: column (col0 + ct*16 + l) of B_n, rows k0+2*half (+1)
  const float* Bcol = B + (size_t)n * KDIM * TV + (size_t)(2 * half) * TV + (col0 + l);

  v8f acc[MT_W][CT_W];
#pragma unroll
  for (int mt = 0; mt < MT_W; ++mt)
#pragma unroll
    for (int ct = 0; ct < CT_W; ++ct) acc[mt][ct] = (v8f){};

  for (int k0 = 0; k0 < KDIM; k0 += 4) {
    v2f a[MT_W], b[CT_W];
#pragma unroll
    for (int mt = 0; mt < MT_W; ++mt) {
      const float* w2 = Wrow + (size_t)mt * 16 * KDIM + k0;  // contiguous pair -> b64
      a[mt].x = w2[0];
      a[mt].y = w2[1];
    }
#pragma unroll
    for (int ct = 0; ct < CT_W; ++ct) {
      b[ct].x = Bcol[(size_t)k0 * TV + ct * 16];
      b[ct].y = Bcol[(size_t)(k0 + 1) * TV + ct * 16];
    }
    // 8 independent WMMAs per 8 loads; accumulator chains are independent so
    // the scheduler can issue them back-to-back over the XDL pipe.
#pragma unroll
    for (int mt = 0; mt < MT_W; ++mt)
#pragma unroll
      for (int ct = 0; ct < CT_W; ++ct)
        acc[mt][ct] = __builtin_amdgcn_wmma_f32_16x16x4_f32(
            /*neg_a=*/false, a[mt], /*neg_b=*/false, b[ct],
            /*c_mod=*/(short)0, acc[mt][ct],
            /*reuse_a=*/false, /*reuse_b=*/false);
  }

  // C/D layout: VGPR j holds row (tile_m0 + j + 8*half), column tile_col0 + l
  const size_t outBase = (size_t)n * EMB * TV;
#pragma unroll
  for (int mt = 0; mt < MT_W; ++mt) {
#pragma unroll
    for (int ct = 0; ct < CT_W; ++ct) {
      const int col = col0 + ct * 16 + l;
#pragma unroll
      for (int j = 0; j < 8; ++j) {
        const int row = m0 + mt * 16 + j + 8 * half;
        const size_t idx = outBase + (size_t)row * TV + col;
        float r = acc[mt][ct][j];
        if (MODE == 0) {
          out[idx] = r + bias[row];
        } else if (MODE == 1) {
          const int t = col / V_;
          out[idx] = (r + bias[row]) * pmul[((size_t)n * EMB + row) * T_ + t];
        } else {
          out[idx] = fmaf(r, INV_WIN, out[idx]);
        }
      }
    }
  }
}

// ---------------------------------------------------------------------------
// scores[n,h,t,v] = (sum_c Wa[h,c]*xx[n,c,t,v] + ba[h]) / sqrt(HC)
// One block per (n,h); threads stride over tv (coalesced).
// ---------------------------------------------------------------------------
__global__ __launch_bounds__(256)
void scores_kernel(const float* __restrict__ xx, const float* __restrict__ Wa,
                   const float* __restrict__ ba, float* __restrict__ scores) {
  const int n = blockIdx.x / HEADS;
  const int h = blockIdx.x % HEADS;
  const float bh = ba[h];
  const float* Wh = Wa + (size_t)h * EMB;
  const float* xn = xx + (size_t)n * EMB * TV;
  float* sn = scores + ((size_t)n * HEADS + h) * TV;
  for (int tv = threadIdx.x; tv < TV; tv += 256) {
    float s = bh;
    for (int c = 0; c < EMB; ++c)
      s = fmaf(Wh[c], xn[(size_t)c * TV + tv], s);
    sn[tv] = s * INV_SQRT_HC;
  }
}

// ---------------------------------------------------------------------------
// Attention pool. One block per (n,t); 8 waves = 8 heads.
// Phase 1: wave h builds softmax over WIN*V=125 gathered scores (invalid
//          positions contribute value 0.0 *before* softmax, as the reference).
// Phase 2: thread c reduces pooled[n,c,t] = sum_{w,v} wsoft[h][w,v]*xx[n,c,sidx,v]
// ---------------------------------------------------------------------------
__global__ __launch_bounds__(256)
void pool_kernel(const float* __restrict__ xx, const float* __restrict__ scores,
                 float* __restrict__ pooled /* [N,EMB,T] */) {
  __shared__ float wsoft[HEADS][WIN * V_ + 3];  // stride 128
  const int n = blockIdx.x / T_;
  const int t = blockIdx.x % T_;
  const int wave = threadIdx.x >> 5;
  const int lane = threadIdx.x & 31;

  {
    const int h = wave;
    const float* sh = scores + ((size_t)n * HEADS + h) * TV;
    float val[4];
    float m = -3.0e38f;
#pragma unroll
    for (int i = 0; i < 4; ++i) {
      const int p = lane + 32 * i;
      float vv = -3.0e38f;
      if (p < WIN * V_) {
        const int w = p / V_, v = p % V_;
        const int tidx = t - PAD + w * DIL;
        vv = (tidx >= 0 && tidx < T_) ? sh[tidx * V_ + v] : 0.0f;
      }
      val[i] = vv;
      m = fmaxf(m, vv);
    }
    for (int off = 16; off > 0; off >>= 1) m = fmaxf(m, __shfl_xor(m, off, 32));
    float s = 0.0f;
#pragma unroll
    for (int i = 0; i < 4; ++i) {
      const int p = lane + 32 * i;
      float e = (p < WIN * V_) ? __expf(val[i] - m) : 0.0f;
      val[i] = e;
      s += e;
    }
    for (int off = 16; off > 0; off >>= 1) s += __shfl_xor(s, off, 32);
    const float inv = 1.0f / s;
#pragma unroll
    for (int i = 0; i < 4; ++i) {
      const int p = lane + 32 * i;
      if (p < WIN * V_) wsoft[h][p] = val[i] * inv;
    }
  }
  __syncthreads();

  const int c = threadIdx.x;          // output channel (h = c/32)
  const int h = c >> 5;
  const float* xc = xx + ((size_t)n * EMB + c) * TV;
  float acc = 0.0f;
#pragma unroll
  for (int w = 0; w < WIN; ++w) {
    const float* xr = xc + shift_t(w, t) * V_;
    const float* ww = &wsoft[h][w * V_];
#pragma unroll
    for (int v = 0; v < V_; ++v) acc = fmaf(ww[v], xr[v], acc);
  }
  pooled[((size_t)n * EMB + c) * T_ + t] = acc;
}

// d_out[n,c,t,v] = (sum_w q[n,c,sidx[w,t],v]) / WIN + bkv[c]   (partial)
__global__ __launch_bounds__(256)
void sq_partial_kernel(const float* __restrict__ q, const float* __restrict__ bkv,
                       float* __restrict__ outp) {
  const size_t idx = (size_t)blockIdx.x * 256 + threadIdx.x;
  const int tv = (int)(idx % TV);
  const size_t ncBase = idx - tv;
  const int c = (int)((idx / TV) % EMB);
  const int t = tv / V_, v = tv % V_;
  float s = 0.0f;
#pragma unroll
  for (int w = 0; w < WIN; ++w) s += q[ncBase + (size_t)shift_t(w, t) * V_ + v];
  outp[idx] = fmaf(s, INV_WIN, bkv[c]);
}

// g[n,c,t,v] = pk[n,c,t] * sum_w v[n,c,sidx[w,t],v]
__global__ __launch_bounds__(256)
void gather_sv_kernel(const float* __restrict__ vbuf, const float* __restrict__ pk,
                      float* __restrict__ g) {
  const size_t idx = (size_t)blockIdx.x * 256 + threadIdx.x;
  const int tv = (int)(idx % TV);
  const size_t ncBase = idx - tv;
  const int t = tv / V_, v = tv % V_;
  float s = 0.0f;
#pragma unroll
  for (int w = 0; w < WIN; ++w) s += vbuf[ncBase + (size_t)shift_t(w, t) * V_ + v];
  g[idx] = s * pk[(idx / TV) * T_ + t];
}

extern "C" void kernel_launch(void* const* d_in, const int* in_sizes, int n_in,
                              void* d_out, int out_size, void* d_ws, size_t ws_size,
                              hipStream_t stream) {
  (void)in_sizes; (void)n_in; (void)out_size; (void)ws_size;
  const float* x   = (const float*)d_in[0];
  const float* Wq  = (const float*)d_in[1];
  const float* bq  = (const float*)d_in[2];
  const float* Wa  = (const float*)d_in[3];
  const float* ba  = (const float*)d_in[4];
  const float* Wk  = (const float*)d_in[5];
  const float* bk  = (const float*)d_in[6];
  const float* Wv  = (const float*)d_in[7];
  const float* bv  = (const float*)d_in[8];
  const float* Wkv = (const float*)d_in[9];
  const float* bkv = (const float*)d_in[10];
  float* out = (float*)d_out;

  const size_t SZ  = (size_t)N_ * EMB * TV;  // 13,107,200 floats (52.4 MB)
  const size_t PSZ = (size_t)N_ * EMB * T_;  //    524,288 floats
  float* bufA = (float*)d_ws;      // q -> kq -> v
  float* bufB = bufA + SZ;         // g (also aliases scores: disjoint lifetimes)
  float* pqb  = bufB + SZ;
  float* pkb  = pqb + PSZ;
  float* scb  = bufB;              // scores alias: last read step 7, bufB first written step 9

  // waves: N * (EMB/64) * (TV/32) = 16*4*100 = 6400 -> 800 blocks of 8 waves
  const int GEMM_GRID = (N_ * (EMB / (16 * MT_W)) * (TV / (16 * CT_W))) / 8;
  const int ELEM_GRID = (int)(SZ / 256);    // 51200

  // 1) q = Wq x + bq
  gemm_wmma_kernel<C_IN, 0><<<GEMM_GRID, 256, 0, stream>>>(Wq, bq, x, nullptr, bufA);
  // 2-3) pq = pool(q)
  scores_kernel<<<N_ * HEADS, 256, 0, stream>>>(bufA, Wa, ba, scb);
  pool_kernel<<<N_ * T_, 256, 0, stream>>>(bufA, scb, pqb);
  // 4) d_out = sq/WIN + bkv (partial)
  sq_partial_kernel<<<ELEM_GRID, 256, 0, stream>>>(bufA, bkv, out);
  // 5) kq = (Wk x + bk) * pq   (pq fused into the GEMM epilogue)
  gemm_wmma_kernel<C_IN, 1><<<GEMM_GRID, 256, 0, stream>>>(Wk, bk, x, pqb, bufA);
  // 6-7) pk = pool(kq)
  scores_kernel<<<N_ * HEADS, 256, 0, stream>>>(bufA, Wa, ba, scb);
  pool_kernel<<<N_ * T_, 256, 0, stream>>>(bufA, scb, pkb);
  // 8) v = Wv x + bv
  gemm_wmma_kernel<C_IN, 0><<<GEMM_GRID, 256, 0, stream>>>(Wv, bv, x, nullptr, bufA);
  // 9) g = pk * sum_w shift(v)
  gather_sv_kernel<<<ELEM_GRID, 256, 0, stream>>>(bufA, pkb, bufB);
  // 10) d_out += (Wkv g)/WIN     (algebraic collapse of the w-sum)
  gemm_wmma_kernel<EMB, 2><<<GEMM_GRID, 256, 0, stream>>>(Wkv, nullptr, bufB, nullptr, out);
}